// GradientTransform_36163624632691
// MI455X (gfx1250) — compile-verified
//
#include <hip/hip_runtime.h>

typedef __attribute__((ext_vector_type(2))) float v2f;
typedef __attribute__((ext_vector_type(8))) float v8f;
typedef __attribute__((ext_vector_type(4))) int   v4i;

namespace {
constexpr int kXD = 1024;   // x_dim
constexpr int kVD = 4096;   // delta_dim
constexpr int kD  = 5120;   // concat dim
constexpr int kN  = 16384;  // samples
constexpr int kR  = 64;     // rank
constexpr float kEps = 1e-7f;

constexpr int kRB      = 32;        // row-block partials for stats
constexpr int kRowsPer = kN / kRB;  // 512 rows per partial

constexpr int kKC   = 64;           // GEMM1 K-chunk staged in LDS
constexpr int kNC   = kD / kKC;     // 80 chunks
constexpr int kAPad = 68;           // padded LDS row pitch (floats): bank = 4*row+k
}

// -------- async global->LDS path (guarded; falls back to register staging) --
#if __has_builtin(__builtin_amdgcn_global_load_async_to_lds_b128) && \
    __has_builtin(__builtin_amdgcn_s_wait_asynccnt)
#define GT_ASYNC_LDS 1
typedef __attribute__((address_space(1))) v4i* gas_v4i_ptr;  // global int4*
typedef __attribute__((address_space(3))) v4i* lds_v4i_ptr;  // LDS int4*

__device__ __forceinline__ gas_v4i_ptr gt_to_gas(const float* p) {
  return (gas_v4i_ptr)(v4i*)const_cast<float*>(p);
}
__device__ __forceinline__ lds_v4i_ptr gt_to_lds(float* p) {
  return (lds_v4i_ptr)(v4i*)p;
}
#else
#define GT_ASYNC_LDS 0
#endif

// ---------------------------------------------------------------------------
// Phase 1a: partial per-column sums over 512-row blocks (640 blocks -> enough
// waves to saturate HBM).  Deterministic two-stage reduction, no atomics.
// ---------------------------------------------------------------------------
__global__ void gt_stats_partial(const float* __restrict__ u,
                                 const float* __restrict__ v,
                                 float* __restrict__ psum,
                                 float* __restrict__ psumsq) {
  constexpr int nColBlk = kD / 256;  // 20
  int colblk = blockIdx.x % nColBlk;
  int rowblk = blockIdx.x / nColBlk;
  int col = colblk * 256 + threadIdx.x;
  const float* src;
  int width, j;
  if (col < kXD) { src = u; width = kXD; j = col; }
  else           { src = v; width = kVD; j = col - kXD; }
  const float* p = src + (size_t)rowblk * kRowsPer * width + j;
  float s = 0.f, s2 = 0.f;
  for (int n = 0; n < kRowsPer; ++n) {
    float x = p[(size_t)n * width];
    s += x;
    s2 = fmaf(x, x, s2);
  }
  psum[rowblk * kD + col]   = s;
  psumsq[rowblk * kD + col] = s2;
}

// ---------------------------------------------------------------------------
// Phase 1b: fold the 32 partials -> mean, 1/(std+eps).
// ---------------------------------------------------------------------------
__global__ void gt_stats_final(const float* __restrict__ psum,
                               const float* __restrict__ psumsq,
                               float* __restrict__ mean,
                               float* __restrict__ rstd) {
  int col = blockIdx.x * blockDim.x + threadIdx.x;
  if (col >= kD) return;
  float s = 0.f, s2 = 0.f;
#pragma unroll 8
  for (int r = 0; r < kRB; ++r) {
    s  += psum[r * kD + col];
    s2 += psumsq[r * kD + col];
  }
  float m = s * (1.0f / kN);
  float var = (s2 - (float)kN * m * m) * (1.0f / (float)(kN - 1));
  var = fmaxf(var, 0.f);
  mean[col] = m;
  rstd[col] = 1.0f / (sqrtf(var) + kEps);
}

// ---------------------------------------------------------------------------
// Phase 2: x = (in - mean) * rstd into d_out (split u/v output layout).
// ---------------------------------------------------------------------------
__global__ void gt_norm_kernel(const float* __restrict__ u,
                               const float* __restrict__ v,
                               const float* __restrict__ mean,
                               const float* __restrict__ rstd,
                               float* __restrict__ xu,
                               float* __restrict__ xv) {
  size_t i0 = (size_t)blockIdx.x * blockDim.x + threadIdx.x;
  size_t stride = (size_t)gridDim.x * blockDim.x;
  size_t nu = (size_t)kN * kXD;
  size_t nv = (size_t)kN * kVD;
  for (size_t k = i0; k < nu; k += stride) {
    int j = (int)(k & (kXD - 1));
    xu[k] = (u[k] - mean[j]) * rstd[j];
  }
  for (size_t k = i0; k < nv; k += stride) {
    int j = (int)(k & (kVD - 1));
    xv[k] = (v[k] - mean[kXD + j]) * rstd[kXD + j];
  }
}

// ---------------------------------------------------------------------------
// GEMM1: t[N,64] = x[N,5120] @ wv^T.  Block = 8 waves covering a 32-row x
// 64-col tile of t (2 m-tiles x 4 r-tiles).  A chunks (32x64 f32) are staged
// into double-buffered LDS via async global->LDS loads so each x row is
// fetched from global exactly once per block (was 4x).  A is consumed with
// conflict-free ds_load_b64 (pitch 68 -> bank = 4*row + k, 64 distinct).
// B (wv rows, 1.25 MB) streams from global and lives in L2 (192 MB).
// ---------------------------------------------------------------------------
__global__ void __launch_bounds__(256)
gt_gemm1_kernel(const float* __restrict__ xu,
                const float* __restrict__ xv,
                const float* __restrict__ wv,  // [64, 5120]
                float* __restrict__ t) {       // [N, 64]
  __shared__ float As[2][32 * kAPad];

  const int tid  = threadIdx.x;
  const int wid  = tid >> 5;
  const int lane = tid & 31;
  const int half = lane >> 4;
  const int l    = lane & 15;

  const int m0blk = blockIdx.x * 32;
  const int mi = wid >> 2;          // 0..1 : which 16-row sub-tile
  const int r0 = (wid & 3) * 16;    // 0,16,32,48

  // staging map: 8 threads per row, 8 floats (2 x b128) per thread
  const int srow = tid >> 3;
  const int scol = (tid & 7) * 8;

  auto stage = [&](int c, int buf) {
    int k0 = c * kKC;
    const float* g;
    if (k0 < kXD) g = xu + (size_t)(m0blk + srow) * kXD + k0 + scol;
    else          g = xv + (size_t)(m0blk + srow) * kVD + (k0 - kXD) + scol;
    float* lp = &As[buf][srow * kAPad + scol];
#if GT_ASYNC_LDS
    __builtin_amdgcn_global_load_async_to_lds_b128(gt_to_gas(g),     gt_to_lds(lp),     0, 0);
    __builtin_amdgcn_global_load_async_to_lds_b128(gt_to_gas(g + 4), gt_to_lds(lp + 4), 0, 0);
#else
    *(float4*)lp       = *(const float4*)g;
    *(float4*)(lp + 4) = *(const float4*)(g + 4);
#endif
  };

  v8f acc = {};
  const float* brow = wv + (size_t)(r0 + l) * kD + 2 * half;

  stage(0, 0);
  for (int c = 0; c < kNC; ++c) {
    if (c + 1 < kNC) {
      stage(c + 1, (c + 1) & 1);       // prefetch next chunk into other buffer
#if GT_ASYNC_LDS
      __builtin_amdgcn_s_wait_asynccnt(2);  // chunk c retired; c+1 in flight
#endif
    } else {
#if GT_ASYNC_LDS
      __builtin_amdgcn_s_wait_asynccnt(0);
#endif
    }
    __syncthreads();                   // chunk c visible to all waves

    const float* a = &As[c & 1][(mi * 16 + l) * kAPad + 2 * half];
    const float* b = brow + c * kKC;
#pragma unroll
    for (int k = 0; k < kKC; k += 4) {
      v2f av = *(const v2f*)(a + k);   // ds_load_b64, bank-conflict free
      v2f bv = *(const v2f*)(b + k);   // global_load_b64, L2-resident panel
      acc = __builtin_amdgcn_wmma_f32_16x16x4_f32(
          false, av, false, bv, (short)0, acc, false, false);
    }
    __syncthreads();                   // buffer free before overwrite at c+2
  }

  // C/D layout: VGPR rr -> row +rr (lanes 0-15) / row +rr+8 (lanes 16-31)
  float* trow = t + (size_t)(m0blk + mi * 16 + 8 * half) * kR + r0 + l;
#pragma unroll
  for (int rr = 0; rr < 8; ++rr)
    trow[(size_t)rr * kR] = acc[rr];
}

// ---------------------------------------------------------------------------
// GEMM2 + epilogue: x[n, j] += relu( t[n,:] . wu[j,:] + b[j] ), in place.
// K = 64 fully unrolled (16 WMMA steps); t (4 MB) stays hot in L2 across the
// 320 j-tiles that reuse each row block.
// ---------------------------------------------------------------------------
__global__ void __launch_bounds__(256)
gt_gemm2_kernel(const float* __restrict__ t,   // [N, 64]
                const float* __restrict__ wu,  // [5120, 64]
                const float* __restrict__ bias,
                float* __restrict__ xu,
                float* __restrict__ xv) {
  int wid  = threadIdx.x >> 5;
  int lane = threadIdx.x & 31;
  int half = lane >> 4;
  int l    = lane & 15;

  int tile  = blockIdx.x * 8 + wid;
  int mtile = tile / (kD / 16);
  int jtile = tile % (kD / 16);
  int m0 = mtile * 16;
  int j0 = jtile * 16;

  v8f acc = {};
  const float* arow = t  + (size_t)(m0 + l) * kR + 2 * half;
  const float* brow = wu + (size_t)(j0 + l) * kR + 2 * half;
#pragma unroll
  for (int k = 0; k < kR; k += 4) {
    v2f a = *(const v2f*)(arow + k);
    v2f b = *(const v2f*)(brow + k);
    acc = __builtin_amdgcn_wmma_f32_16x16x4_f32(
        false, a, false, b, (short)0, acc, false, false);
  }

  int col = j0 + l;                    // 16-wide tile stays in one region
  float bv = bias[col];
  float* xcol = (col < kXD) ? (xu + col) : (xv + (col - kXD));
  int ld = (col < kXD) ? kXD : kVD;
#pragma unroll
  for (int rr = 0; rr < 8; ++rr) {
    int row = m0 + rr + 8 * half;
    float pre = fmaxf(acc[rr] + bv, 0.f);
    float* p = xcol + (size_t)row * ld;
    *p += pre;                         // residual: relu(pre) + x
  }
}

// ---------------------------------------------------------------------------
extern "C" void kernel_launch(void* const* d_in, const int* in_sizes, int n_in,
                              void* d_out, int out_size, void* d_ws, size_t ws_size,
                              hipStream_t stream) {
  (void)in_sizes; (void)n_in; (void)out_size; (void)ws_size;

  const float* u   = (const float*)d_in[0];
  const float* v   = (const float*)d_in[1];
  const float* wu0 = (const float*)d_in[2];
  const float* wv0 = (const float*)d_in[3];
  const float* b0  = (const float*)d_in[4];
  const float* wu1 = (const float*)d_in[5];
  const float* wv1 = (const float*)d_in[6];
  const float* b1  = (const float*)d_in[7];

  float* mean   = (float*)d_ws;                 // [5120]
  float* rstd   = mean + kD;                    // [5120]
  float* t      = rstd + kD;                    // [16384 * 64]
  float* psum   = t + (size_t)kN * kR;          // [32 * 5120]
  float* psumsq = psum + kRB * kD;              // [32 * 5120]  (~5.5 MB total)

  float* xu = (float*)d_out;                    // [N, 1024] (output layout)
  float* xv = xu + (size_t)kN * kXD;            // [N, 4096]

  gt_stats_partial<<<kRB * (kD / 256), 256, 0, stream>>>(u, v, psum, psumsq);
  gt_stats_final<<<kD / 256, 256, 0, stream>>>(psum, psumsq, mean, rstd);
  gt_norm_kernel<<<4096, 256, 0, stream>>>(u, v, mean, rstd, xu, xv);

  const int g1 = kN / 32;                       // 512 blocks, 8 waves each
  const int g2 = (kN / 16) * (kD / 16) / 8;     // 40960 blocks

  gt_gemm1_kernel<<<g1, 256, 0, stream>>>(xu, xv, wv0, t);
  gt_gemm2_kernel<<<g2, 256, 0, stream>>>(t, wu0, b0, xu, xv);
  gt_gemm1_kernel<<<g1, 256, 0, stream>>>(xu, xv, wv1, t);
  gt_gemm2_kernel<<<g2, 256, 0, stream>>>(t, wu1, b1, xu, xv);
}